// STGCN_81449759802098
// MI455X (gfx1250) — compile-verified
//
#include <hip/hip_runtime.h>
#include <hip/hip_bf16.h>

// B=2048, N=60, T=5, C=5, F_OUT=300, K=3
// Workspace layout (floats):
//   Abuf : 51200*60            (graph-conv A matrix, rows = b*25+ct, cols = n)
//   WbufT: 3*304*64            (fused Chebyshev weights, TRANSPOSED: [hop][f][k],
//                               f padded 300->304, k padded 60->64, zero-filled)
//   obuf : 2048*300*25         (graph-conv out in (b, f, c*5+t) layout == h2 layout)
//   bn   : 120                 (mu[60], rsqrt(var+eps)[60])
// Total ~74 MB.

typedef __attribute__((ext_vector_type(2))) float v2f;
typedef __attribute__((ext_vector_type(8))) float v8f;

#define NB 2048
#define NN 60
#define NT 5
#define NC 5
#define NF 300
#define NFP 304              // padded F
#define NKP 64               // padded K (graph dimension) for aligned b64 B loads
#define NR (NB * 25)         // 51200 GEMM rows
#define OB_PER_B (NF * 25)   // 7500
#define WT_SZ (NFP * NKP)    // one transposed weight matrix

// ---------------------------------------------------------------------------
// Kernel 1: temporal attention (rank-1 factored) + TCN1, one block per batch.
// ---------------------------------------------------------------------------
__global__ __launch_bounds__(64) void k_attn_tcn1(
    const float* __restrict__ X,  const float* __restrict__ U1,
    const float* __restrict__ U2, const float* __restrict__ U3,
    const float* __restrict__ be, const float* __restrict__ Ve,
    const float* __restrict__ w0, const float* __restrict__ b0,
    const float* __restrict__ dw, const float* __restrict__ db,
    const float* __restrict__ ws, const float* __restrict__ bs,
    float* __restrict__ Abuf)
{
  __shared__ float xs[NN][NT];
  __shared__ float lhs1[NT], q[NT];
  __shared__ float Em[NT][NT];
  __shared__ float att[NT][NT];
  __shared__ float lw0[15], lb0[5], ldw[5], ldb[5], lws[300], lbs[20];

  const int b = blockIdx.x;
  const int tid = threadIdx.x;

  for (int i = tid; i < 15;  i += 64) lw0[i] = w0[i];
  for (int i = tid; i < 5;   i += 64) { lb0[i] = b0[i]; ldw[i] = dw[i]; ldb[i] = db[i]; }
  for (int i = tid; i < 300; i += 64) lws[i] = ws[i];
  for (int i = tid; i < 20;  i += 64) lbs[i] = bs[i];

  if (tid < NN)
    for (int t = 0; t < NT; ++t) xs[tid][t] = X[(b * NN + tid) * NT + t];
  __syncthreads();

  // lhs1[t] = sum_n X[b,n,t]*U1[n];  q[t] = sum_n U2[n]*X[b,n,t]
  if (tid < 10) {
    int t = tid % 5; float s = 0.f;
    if (tid < 5) { for (int n = 0; n < NN; ++n) s += xs[n][t] * U1[n]; lhs1[t] = s; }
    else         { for (int n = 0; n < NN; ++n) s += U2[n] * xs[n][t]; q[t]    = s; }
  }
  __syncthreads();

  // E = Ve @ sigmoid(lhs1[s]*U3*q[t'] + be[s,t'])
  if (tid < 25) {
    int t = tid / 5, tp = tid % 5;
    float u3 = U3[0], e = 0.f;
    for (int s = 0; s < 5; ++s) {
      float p = lhs1[s] * u3 * q[tp] + be[s * 5 + tp];
      e += Ve[t * 5 + s] * (1.f / (1.f + __expf(-p)));
    }
    Em[t][tp] = e;
  }
  __syncthreads();

  // column softmax (over axis 1 == t)
  if (tid < 5) {
    int tp = tid;
    float m = Em[0][tp];
    for (int t = 1; t < 5; ++t) m = fmaxf(m, Em[t][tp]);
    float ex[5], ssum = 0.f;
    for (int t = 0; t < 5; ++t) { ex[t] = __expf(Em[t][tp] - m); ssum += ex[t]; }
    for (int t = 0; t < 5; ++t) att[t][tp] = ex[t] / ssum;
  }
  __syncthreads();

  if (tid < NN) {
    const int n = tid;
    float x[5];
    for (int tp = 0; tp < 5; ++tp) {
      float s = 0.f;
      for (int t = 0; t < 5; ++t) s += xs[n][t] * att[t][tp];
      x[tp] = s;
    }
    // ---- TCN1 in registers ----
    float h[5][5];
    for (int c = 0; c < 5; ++c)
      for (int t = 0; t < 5; ++t) {
        float r = ldw[c] * x[t] + ldb[c];                 // 1x1 residual proj
        float a = lb0[c];
        for (int j = 0; j < 3; ++j) { int tt = t - (2 - j); if (tt >= 0) a += lw0[c * 3 + j] * x[tt]; }
        h[c][t] = fmaxf(fmaxf(a, 0.f) + r, 0.f);
      }
    for (int i = 0; i < 4; ++i) {
      int d = 2 << i;                                     // 2,4,8,16
      float nh[5][5];
      for (int c = 0; c < 5; ++c)
        for (int t = 0; t < 5; ++t) {
          float a = lbs[i * 5 + c];
          for (int ci = 0; ci < 5; ++ci)
            for (int j = 0; j < 3; ++j) {
              int tt = t - (2 - j) * d;
              if (tt >= 0) a += lws[((i * 5 + c) * 5 + ci) * 3 + j] * h[ci][tt];
            }
          nh[c][t] = a;
        }
      for (int c = 0; c < 5; ++c)
        for (int t = 0; t < 5; ++t)
          h[c][t] = fmaxf(fmaxf(nh[c][t], 0.f) + h[c][t], 0.f);
    }
    for (int c = 0; c < 5; ++c)
      for (int t = 0; t < 5; ++t)
        Abuf[(b * 25 + c * 5 + t) * NN + n] = h[c][t];
  }
}

// ---------------------------------------------------------------------------
// Kernels 2/3: fused Chebyshev weights, stored TRANSPOSED WT[hop][f][k]:
//   W0=Theta, W1=A^T Theta, W2=2 A^T W1 - Theta   (T2 = 2A^2 - I)
// ---------------------------------------------------------------------------
__global__ void k_w01(const float* __restrict__ A, const float* __restrict__ Th,
                      float* __restrict__ WT) {
  int idx = blockIdx.x * blockDim.x + threadIdx.x;
  if (idx >= WT_SZ) return;
  int f = idx / NKP, n = idx % NKP;   // n is the K (graph) index
  float w0 = 0.f, w1 = 0.f;
  if (f < NF && n < NN) {
    w0 = Th[n * NF + f];
    for (int m = 0; m < NN; ++m) w1 += A[m * NN + n] * Th[m * NF + f];
  }
  WT[idx] = w0;
  WT[WT_SZ + idx] = w1;
}

__global__ void k_w2(const float* __restrict__ A, const float* __restrict__ Th,
                     float* __restrict__ WT) {
  int idx = blockIdx.x * blockDim.x + threadIdx.x;
  if (idx >= WT_SZ) return;
  int f = idx / NKP, n = idx % NKP;
  float w2 = 0.f;
  if (f < NF && n < NN) {
    float s = 0.f;
    for (int qq = 0; qq < NN; ++qq) s += A[qq * NN + n] * WT[WT_SZ + f * NKP + qq];
    w2 = 2.f * s - Th[n * NF + f];
  }
  WT[2 * WT_SZ + idx] = w2;
}

// ---------------------------------------------------------------------------
// Kernel 4: Chebyshev graph conv GEMM via V_WMMA_F32_16X16X4_F32.
// One wave per 16x16 output tile; A fragment shared across the 3 hop weights;
// B fragments are single aligned b64 loads thanks to the transposed W layout.
// out(b,f,ct) = relu( sum_k relu( sum_n A[b*25+ct, n] * Wk[n,f] ) )
// ---------------------------------------------------------------------------
__global__ __launch_bounds__(32) void k_cheby(const float* __restrict__ Abuf,
                                              const float* __restrict__ WT,
                                              float* __restrict__ obuf) {
  const int mt = blockIdx.x;            // 0..3199
  const int ft = blockIdx.y;            // 0..18
  const int l = threadIdx.x;
  const int m16 = l & 15;
  const int koff = (l >> 4) * 2;        // K sub-offset per half-wave (ISA A/B layout)
  const int rbase = mt * 16;
  const int fbase = ft * 16;

  const float* __restrict__ arow = Abuf + (size_t)(rbase + m16) * NN;
  const float* __restrict__ w0r = WT + (size_t)(fbase + m16) * NKP;   // WT[0][f][*]
  const float* __restrict__ w1r = w0r + WT_SZ;
  const float* __restrict__ w2r = w0r + 2 * WT_SZ;

  v8f acc0 = {}; v8f acc1 = {}; v8f acc2 = {};
  for (int k0 = 0; k0 < NN; k0 += 4) {
    const int ka = k0 + koff;           // even -> 8B aligned
    v2f a   = *(const v2f*)(arow + ka);
    v2f b0v = *(const v2f*)(w0r + ka);
    v2f b1v = *(const v2f*)(w1r + ka);
    v2f b2v = *(const v2f*)(w2r + ka);
    acc0 = __builtin_amdgcn_wmma_f32_16x16x4_f32(false, a, false, b0v, (short)0, acc0, false, false);
    acc1 = __builtin_amdgcn_wmma_f32_16x16x4_f32(false, a, false, b1v, (short)0, acc1, false, false);
    acc2 = __builtin_amdgcn_wmma_f32_16x16x4_f32(false, a, false, b2v, (short)0, acc2, false, false);
  }
  const int f = fbase + m16;
  for (int v = 0; v < 8; ++v) {
    float r = fmaxf(acc0[v], 0.f) + fmaxf(acc1[v], 0.f) + fmaxf(acc2[v], 0.f);
    r = fmaxf(r, 0.f);
    int rr = rbase + v + (l >> 4) * 8;  // C layout: lanes16-31 hold M = v+8
    if (f < NF) {
      int bb = rr / 25, ct = rr % 25;
      obuf[bb * OB_PER_B + f * 25 + ct] = r;
    }
  }
}

// ---------------------------------------------------------------------------
// Kernel 5: TCN2 — one thread per (b,f) row, 5 dilated layers in registers.
// ---------------------------------------------------------------------------
__global__ __launch_bounds__(256) void k_tcn2(float* __restrict__ obuf,
                                              const float* __restrict__ ws,
                                              const float* __restrict__ bs) {
  __shared__ float lw[375], lb[25];
  const int tid = threadIdx.x;
  for (int i = tid; i < 375; i += 256) lw[i] = ws[i];
  for (int i = tid; i < 25;  i += 256) lb[i] = bs[i];
  __syncthreads();
  const int gid = blockIdx.x * 256 + tid;
  if (gid >= NB * NF) return;
  float* row = obuf + (size_t)gid * 25;
  float h[5][5];
  for (int c = 0; c < 5; ++c) for (int t = 0; t < 5; ++t) h[c][t] = row[c * 5 + t];
  for (int i = 0; i < 5; ++i) {
    int d = 1 << i;                                       // 1,2,4,8,16
    float nh[5][5];
    for (int c = 0; c < 5; ++c)
      for (int t = 0; t < 5; ++t) {
        float a = lb[i * 5 + c];
        for (int ci = 0; ci < 5; ++ci)
          for (int j = 0; j < 3; ++j) {
            int tt = t - (2 - j) * d;
            if (tt >= 0) a += lw[((i * 5 + c) * 5 + ci) * 3 + j] * h[ci][tt];
          }
        nh[c][t] = a;
      }
    for (int c = 0; c < 5; ++c)
      for (int t = 0; t < 5; ++t)
        h[c][t] = fmaxf(fmaxf(nh[c][t], 0.f) + h[c][t], 0.f);
  }
  for (int c = 0; c < 5; ++c) for (int t = 0; t < 5; ++t) row[c * 5 + t] = h[c][t];
}

// ---------------------------------------------------------------------------
// Kernel 6: BatchNorm statistics, one block per node n.
// ---------------------------------------------------------------------------
__global__ __launch_bounds__(256) void k_bnstats(const float* __restrict__ obuf,
                                                 float* __restrict__ bn) {
  __shared__ float s1[256], s2[256];
  const int n = blockIdx.x, tid = threadIdx.x;
  float a = 0.f, b2 = 0.f;
  for (int idx = tid; idx < NB * 125; idx += 256) {
    int bb = idx / 125, e = idx % 125;
    float v = obuf[bb * OB_PER_B + n * 125 + e];
    a += v; b2 += v * v;
  }
  s1[tid] = a; s2[tid] = b2; __syncthreads();
  for (int off = 128; off > 0; off >>= 1) {
    if (tid < off) { s1[tid] += s1[tid + off]; s2[tid] += s2[tid + off]; }
    __syncthreads();
  }
  if (tid == 0) {
    float cnt = (float)(NB * 125);
    float mu = s1[0] / cnt;
    float var = s2[0] / cnt - mu * mu;
    bn[n] = mu;
    bn[NN + n] = rsqrtf(var + 1e-5f);
  }
}

// ---------------------------------------------------------------------------
// Kernel 7: BN-normalize + Conv2d(60->60,(5,10)) as im2col WMMA GEMM + FC.
// One block (4 waves) per batch; y[b] staged normalized in LDS; each wave owns
// a 16-wide output-channel tile; M dim = 16 conv output positions (w).
// ---------------------------------------------------------------------------
__global__ __launch_bounds__(128) void k_conv_fc(
    const float* __restrict__ obuf, const float* __restrict__ bn,
    const float* __restrict__ gamma, const float* __restrict__ beta,
    const float* __restrict__ cw, const float* __restrict__ cb,
    const float* __restrict__ fcw, const float* __restrict__ fcb,
    float* __restrict__ out) {
  __shared__ float ylds[OB_PER_B];     // 30 KB
  __shared__ int offtab[3000];         // 12 KB: k -> i*125+kh*25+kw
  const int b = blockIdx.x;
  const int tid = threadIdx.x;

  for (int e = tid; e < OB_PER_B; e += 128) {
    int n = e / 125;
    float v = obuf[(size_t)b * OB_PER_B + e];
    ylds[e] = gamma[n] * (v - bn[n]) * bn[NN + n] + beta[n];
  }
  for (int k = tid; k < 3000; k += 128) {
    int i = k / 50, rem = k % 50;
    offtab[k] = i * 125 + (rem / 10) * 25 + (rem % 10);
  }
  __syncthreads();

  const int wave = tid >> 5;           // 0..3 -> o tiles 0..63
  const int l = tid & 31;
  const int lane16 = l & 15;
  const int koff = (l >> 4) * 2;
  const int obase = wave * 16;
  const int ocol = obase + lane16;
  const int oc = ocol < NN ? ocol : NN - 1;   // safe clamp; junk columns discarded
  const float* __restrict__ cwrow = cw + (size_t)oc * 3000;

  v8f acc = {};
  for (int k0 = 0; k0 < 3000; k0 += 4) {
    const int k = k0 + koff;                    // even -> 8B aligned
    v2f a;  a.x = ylds[offtab[k] + lane16];     // M dim = conv output position w
    a.y = ylds[offtab[k + 1] + lane16];
    v2f bv = *(const v2f*)(cwrow + k);
    acc = __builtin_amdgcn_wmma_f32_16x16x4_f32(false, a, false, bv, (short)0, acc, false, false);
  }
  // bias + relu + FC over the M (=w) dimension, cross-half reduce.
  const float cbv = cb[oc];
  float partial = 0.f;
  for (int v = 0; v < 8; ++v) {
    int mm = v + (l >> 4) * 8;                  // w index 0..15
    float z = fmaxf(acc[v] + cbv, 0.f);
    partial += z * fcw[mm];
  }
  partial += __shfl_xor(partial, 16, 32);
  if (l < 16 && ocol < NN)
    out[b * NN + ocol] = partial + fcb[0];
}

// ---------------------------------------------------------------------------
extern "C" void kernel_launch(void* const* d_in, const int* in_sizes, int n_in,
                              void* d_out, int out_size, void* d_ws, size_t ws_size,
                              hipStream_t stream) {
  const float* A_hat  = (const float*)d_in[0];
  const float* X      = (const float*)d_in[1];
  const float* U1     = (const float*)d_in[2];
  const float* U2     = (const float*)d_in[3];
  const float* U3     = (const float*)d_in[4];
  const float* b_e    = (const float*)d_in[5];
  const float* V_e    = (const float*)d_in[6];
  const float* Theta1 = (const float*)d_in[7];
  const float* t1w0   = (const float*)d_in[8];
  const float* t1b0   = (const float*)d_in[9];
  const float* t1dw   = (const float*)d_in[10];
  const float* t1db   = (const float*)d_in[11];
  const float* t1ws   = (const float*)d_in[12];
  const float* t1bs   = (const float*)d_in[13];
  const float* t2ws   = (const float*)d_in[14];
  const float* t2bs   = (const float*)d_in[15];
  const float* gamma  = (const float*)d_in[16];
  const float* beta   = (const float*)d_in[17];
  const float* conv_w = (const float*)d_in[18];
  const float* conv_b = (const float*)d_in[19];
  const float* fc_w   = (const float*)d_in[20];
  const float* fc_b   = (const float*)d_in[21];
  float* out = (float*)d_out;

  float* Abuf  = (float*)d_ws;                      // 51200*60
  float* WbufT = Abuf + (size_t)NR * NN;            // 3*304*64
  float* obuf  = WbufT + (size_t)3 * WT_SZ;         // 2048*300*25
  float* bnb   = obuf + (size_t)NB * OB_PER_B;      // 120

  k_attn_tcn1<<<NB, 64, 0, stream>>>(X, U1, U2, U3, b_e, V_e,
                                     t1w0, t1b0, t1dw, t1db, t1ws, t1bs, Abuf);
  int wgrid = (WT_SZ + 255) / 256;
  k_w01<<<wgrid, 256, 0, stream>>>(A_hat, Theta1, WbufT);
  k_w2 <<<wgrid, 256, 0, stream>>>(A_hat, Theta1, WbufT);
  k_cheby<<<dim3(NR / 16, (NF + 15) / 16), 32, 0, stream>>>(Abuf, WbufT, obuf);
  k_tcn2<<<(NB * NF + 255) / 256, 256, 0, stream>>>(obuf, t2ws, t2bs);
  k_bnstats<<<NN, 256, 0, stream>>>(obuf, bnb);
  k_conv_fc<<<NB, 128, 0, stream>>>(obuf, bnb, gamma, beta,
                                    conv_w, conv_b, fc_w, fc_b, out);
}